// MyLSTM_29111288332556
// MI455X (gfx1250) — compile-verified
//
#include <hip/hip_runtime.h>

// 3-layer LSTM (T=256,B=256,I=128,H=512) + linear head, fp32, on gfx1250.
//
// Time budget: ~722 GFLOP total, but only the h_{t-1} @ W_hh^T recurrence
// (768 dependent steps) is serial. Plan:
//   (a) xg = input @ W_ih^T + b  -> massively parallel WMMA GEMM, full device
//   (b) serial kernel: 16 WGs own 16 batch rows each; h/c live in LDS;
//       per step only K=512 recurrent GEMM via V_WMMA_F32_16X16X4_F32,
//       k-outer loop with 8 accumulators so one LDS A-load feeds 8 WMMAs
//       while W_hh streams from the 192MB L2 (weights ~13MB total).
//   (c) output linear on last 32 steps, same WMMA tiling.

typedef __attribute__((ext_vector_type(2))) float v2f;
typedef __attribute__((ext_vector_type(8))) float v8f;

#define TT    256
#define BATCH 256
#define HH    512
#define G4H   2048  // 4*H
#define OUTN  128
#define WINN  32

__device__ __forceinline__ v8f wmma_f32_k4(v2f a, v2f b, v8f c) {
  // (neg_a, A, neg_b, B, c_mod, C, reuse_a, reuse_b)
  return __builtin_amdgcn_wmma_f32_16x16x4_f32(false, a, false, b, (short)0, c,
                                               false, false);
}

__device__ __forceinline__ float sigmoidf_fast(float x) {
  return 1.0f / (1.0f + __expf(-x));
}

// ---------------------------------------------------------------------------
// xg[m, 0:2048] = in_act[m, 0:K] @ W^T + bias   (m = t*B + b, M = 65536)
// Grid: (M/16, 2048/128); block 256 threads = 8 waves; wave w owns a 16x16
// tile at col blockIdx.y*128 + w*16. A tile staged in LDS (16 x K).
// ---------------------------------------------------------------------------
__global__ void xg_gemm_kernel(const float* __restrict__ in_act, int K_in,
                               const float* __restrict__ w_ih,
                               const float* __restrict__ bias,
                               float* __restrict__ xg) {
  extern __shared__ float asmem[];  // 16 x K_in (<= 32KB)

  const int tid   = threadIdx.x;
  const int wave  = tid >> 5;
  const int lane  = tid & 31;
  const int half  = lane >> 4;
  const int lrow  = lane & 15;
  const int mbase = blockIdx.x * 16;
  const int col   = blockIdx.y * 128 + wave * 16 + lrow;

  for (int e = tid; e < 16 * K_in; e += 256) {
    int r = e / K_in;
    int c = e - r * K_in;
    asmem[e] = in_act[(size_t)(mbase + r) * K_in + c];
  }
  __syncthreads();

  v8f acc;
  const float bv = bias[col];
#pragma unroll
  for (int r = 0; r < 8; ++r) acc[r] = bv;

  const float* ar = asmem + lrow * K_in + 2 * half;
  const float* br = w_ih + (size_t)col * K_in + 2 * half;
  for (int k = 0; k < K_in; k += 4) {
    v2f a = *(const v2f*)(ar + k);
    v2f b = *(const v2f*)(br + k);
    acc = wmma_f32_k4(a, b, acc);
  }

#pragma unroll
  for (int r = 0; r < 8; ++r) {
    xg[(size_t)(mbase + r + 8 * half) * G4H + col] = acc[r];
  }
}

// ---------------------------------------------------------------------------
// Serial recurrence for one layer. Grid: 16 blocks (16 batch rows each),
// 512 threads = 16 waves; wave w owns gate columns [128w, 128w+128).
// LDS (floats): hsl[16x512]@0, cs[16x512]@8192, gs[16x2048]@16384
//   -> 49152 floats = 196608 B (< 320KB WGP LDS).
// ---------------------------------------------------------------------------
__global__ void lstm_recurrent_kernel(const float* __restrict__ xg,
                                      const float* __restrict__ w_hh,
                                      float* __restrict__ h_out) {
  extern __shared__ float smem[];
  float* hsl = smem;           // h_{t-1} tile
  float* cs  = smem + 8192;    // c tile
  float* gs  = smem + 16384;   // activated gates tile

  const int tid   = threadIdx.x;
  const int wave  = tid >> 5;
  const int lane  = tid & 31;
  const int half  = lane >> 4;
  const int lrow  = lane & 15;
  const int bbase = blockIdx.x * 16;
  const int gate  = wave >> 2;  // 0=i,1=f,2=g,3=o

  for (int e = tid; e < 16 * HH; e += 512) { hsl[e] = 0.0f; cs[e] = 0.0f; }
  __syncthreads();

  // Per-wave invariant pointers: A from LDS h, B base for this wave's 128
  // weight rows (per-tile delta 16*HH floats = 32KB fits the imm offset).
  const float* hr = hsl + lrow * HH + 2 * half;
  const float* wb = w_hh + ((size_t)wave * 128 + lrow) * HH + 2 * half;
  __builtin_prefetch(wb, 0, 0);

  for (int t = 0; t < TT; ++t) {
    const float* xg_t = xg + ((size_t)t * BATCH + bbase) * G4H;

    // 8 accumulators: init from precomputed input contribution (bias folded).
    v8f acc[8];
#pragma unroll
    for (int tile = 0; tile < 8; ++tile) {
#pragma unroll
      for (int r = 0; r < 8; ++r) {
        acc[tile][r] =
            xg_t[(size_t)(r + 8 * half) * G4H + wave * 128 + tile * 16 + lrow];
      }
    }

    // Recurrent GEMM: one A-load feeds 8 WMMAs per k-step.
    for (int k = 0; k < HH; k += 4) {
      v2f a = *(const v2f*)(hr + k);
#pragma unroll
      for (int tile = 0; tile < 8; ++tile) {
        v2f b = *(const v2f*)(wb + (size_t)tile * 16 * HH + k);
        acc[tile] = wmma_f32_k4(a, b, acc[tile]);
      }
    }

    // Activations (gate type uniform per wave) -> LDS gate buffer.
#pragma unroll
    for (int tile = 0; tile < 8; ++tile) {
      const int colbase = wave * 128 + tile * 16;
#pragma unroll
      for (int r = 0; r < 8; ++r) {
        const int m = r + 8 * half;
        float v = acc[tile][r];
        v = (gate == 2) ? tanhf(v) : sigmoidf_fast(v);
        gs[m * G4H + colbase + lrow] = v;
      }
    }
    __syncthreads();

    // Cell/hidden update; h goes back to LDS and out to global.
    for (int e = tid; e < 16 * HH; e += 512) {
      const int bb = e >> 9;
      const int j  = e & (HH - 1);
      const float ig = gs[bb * G4H + j];
      const float fg = gs[bb * G4H + HH + j];
      const float gg = gs[bb * G4H + 2 * HH + j];
      const float og = gs[bb * G4H + 3 * HH + j];
      const float c  = fg * cs[e] + ig * gg;
      const float h  = og * tanhf(c);
      cs[e]  = c;
      hsl[e] = h;
      h_out[((size_t)t * BATCH + bbase + bb) * HH + j] = h;
    }
    __syncthreads();
  }
}

// ---------------------------------------------------------------------------
// y[t,b,o] = hs[T-WIN+t, b, :] . w_out[o,:] + b_out[o]
// Grid: 512 blocks x 256 threads (8 waves; wave w owns cols [16w,16w+16)).
// ---------------------------------------------------------------------------
__global__ void out_linear_kernel(const float* __restrict__ hs,
                                  const float* __restrict__ w_out,
                                  const float* __restrict__ b_out,
                                  float* __restrict__ out) {
  const int tid   = threadIdx.x;
  const int wave  = tid >> 5;
  const int lane  = tid & 31;
  const int half  = lane >> 4;
  const int lrow  = lane & 15;
  const int mbase = blockIdx.x * 16;
  const int nbase = wave * 16;

  v8f acc;
  const float bv = b_out[nbase + lrow];
#pragma unroll
  for (int r = 0; r < 8; ++r) acc[r] = bv;

  const int arow    = mbase + lrow;
  const int t_local = arow >> 8;  // / 256
  const int bb      = arow & 255;
  const float* ar = hs + ((size_t)(t_local + (TT - WINN)) * BATCH + bb) * HH +
                    2 * half;
  const float* br = w_out + (size_t)(nbase + lrow) * HH + 2 * half;

  for (int k = 0; k < HH; k += 4) {
    v2f a = *(const v2f*)(ar + k);
    v2f b = *(const v2f*)(br + k);
    acc = wmma_f32_k4(a, b, acc);
  }

#pragma unroll
  for (int r = 0; r < 8; ++r) {
    const int m = mbase + r + 8 * half;
    out[(size_t)m * OUTN + nbase + lrow] = acc[r];
  }
}

extern "C" void kernel_launch(void* const* d_in, const int* in_sizes, int n_in,
                              void* d_out, int out_size, void* d_ws,
                              size_t ws_size, hipStream_t stream) {
  const float* x      = (const float*)d_in[0];
  const float* w_ih_0 = (const float*)d_in[1];
  const float* w_hh_0 = (const float*)d_in[2];
  const float* b_0    = (const float*)d_in[3];
  const float* w_ih_1 = (const float*)d_in[4];
  const float* w_hh_1 = (const float*)d_in[5];
  const float* b_1    = (const float*)d_in[6];
  const float* w_ih_2 = (const float*)d_in[7];
  const float* w_hh_2 = (const float*)d_in[8];
  const float* b_2    = (const float*)d_in[9];
  const float* w_out  = (const float*)d_in[10];
  const float* b_out  = (const float*)d_in[11];
  float* out = (float*)d_out;

  // Workspace: hA, hB = [T,B,H] (128MB each), xg = [T,B,4H] (512MB).
  const size_t HElems = (size_t)TT * BATCH * HH;  // 33,554,432
  float* hA = (float*)d_ws;
  float* hB = hA + HElems;
  float* xgb = hB + HElems;

  const size_t smemA = (size_t)16 * HH * sizeof(float);    // 32KB (max K)
  const size_t smemR = (size_t)49152 * sizeof(float);      // 196608 B
  (void)hipFuncSetAttribute((const void*)lstm_recurrent_kernel,
                            hipFuncAttributeMaxDynamicSharedMemorySize,
                            (int)smemR);
  (void)hipFuncSetAttribute((const void*)xg_gemm_kernel,
                            hipFuncAttributeMaxDynamicSharedMemorySize,
                            (int)smemA);

  const dim3 xgrid((TT * BATCH) / 16, G4H / 128);  // 4096 x 16 blocks

  // Layer 0: input x (K=128) -> hA
  xg_gemm_kernel<<<xgrid, 256, 16 * 128 * sizeof(float), stream>>>(
      x, 128, w_ih_0, b_0, xgb);
  lstm_recurrent_kernel<<<16, 512, smemR, stream>>>(xgb, w_hh_0, hA);

  // Layer 1: hA (K=512) -> hB
  xg_gemm_kernel<<<xgrid, 256, smemA, stream>>>(hA, 512, w_ih_1, b_1, xgb);
  lstm_recurrent_kernel<<<16, 512, smemR, stream>>>(xgb, w_hh_1, hB);

  // Layer 2: hB (K=512) -> hA
  xg_gemm_kernel<<<xgrid, 256, smemA, stream>>>(hB, 512, w_ih_2, b_2, xgb);
  lstm_recurrent_kernel<<<16, 512, smemR, stream>>>(xgb, w_hh_2, hA);

  // Output head on last WIN timesteps.
  out_linear_kernel<<<(WINN * BATCH) / 16, 256, 0, stream>>>(hA, w_out, b_out,
                                                             out);
}